// MultiLabelSoftmaxWithCrossEntropyLoss_75900662054989
// MI455X (gfx1250) — compile-verified
//
#include <hip/hip_runtime.h>
#include <hip/hip_bf16.h>

#define BLOCK 256
#define C_DIM 8192
#define P_DIM 8
#define EPT (C_DIM / BLOCK) /* 32 elements per thread */

typedef __attribute__((ext_vector_type(2))) float v2f;
typedef __attribute__((ext_vector_type(8))) float v8f;
typedef int vb128 __attribute__((vector_size(16))); /* <4 x i32> payload */

#if defined(__has_builtin)
#if __has_builtin(__builtin_amdgcn_global_load_async_to_lds_b128)
#define USE_ASYNC_LDS 1
#endif
#if __has_builtin(__builtin_amdgcn_s_wait_asynccnt)
#define HAVE_WAIT_ASYNC_BUILTIN 1
#endif
#endif

// Exact fp32 sum of one value per lane across a wave32, using
// V_WMMA_F32_16X16X4_F32 with B = all-ones. D[m][n] = sum_k A[m][k], so the
// total of all A entries (== the 32 per-lane inputs) is recovered by summing
// a lane's 8 accumulator registers (covers M=0..7 or M=8..15 per half) and
// adding the opposite half via one shfl_xor(16). Exact: every product is x*1.0
// with fp32 accumulation.
__device__ __forceinline__ float wave_sum32_wmma(float x) {
  v2f a;
  a.x = x;
  a.y = 0.0f;
  v2f ones;
  ones.x = 1.0f;
  ones.y = 1.0f;
  v8f c = {};
  c = __builtin_amdgcn_wmma_f32_16x16x4_f32(false, a, false, ones, (short)0, c,
                                            false, false);
  float s = c[0] + c[1] + c[2] + c[3] + c[4] + c[5] + c[6] + c[7];
  s += __shfl_xor(s, 16, 32);
  return s;
}

__device__ __forceinline__ float wave_max32(float x) {
#pragma unroll
  for (int off = 16; off >= 1; off >>= 1)
    x = fmaxf(x, __shfl_xor(x, off, 32));
  return x;
}

__global__ void __launch_bounds__(BLOCK)
mlce_row_kernel(const float* __restrict__ pred,
                const long long* __restrict__ labels,
                float* __restrict__ row_loss) {
  __shared__ float rowbuf[C_DIM]; // 32 KB: the whole row, read from HBM once
  __shared__ float wred[8];
  __shared__ float bcast[2];
  __shared__ float pose[P_DIM];
  __shared__ float posx[P_DIM];

  const int tid = threadIdx.x;
  const int b = blockIdx.x;
  const float* rp = pred + (size_t)b * C_DIM;

  // ---- Stage row into LDS (CDNA5 async global->LDS path when available) ----
#ifdef USE_ASYNC_LDS
#pragma unroll
  for (int i = 0; i < EPT / 4; ++i) {
    const int idx = (i * BLOCK + tid) * 4;
    __builtin_amdgcn_global_load_async_to_lds_b128(
        (__attribute__((address_space(1))) vb128*)(rp + idx),
        (__attribute__((address_space(3))) vb128*)(&rowbuf[idx]), 0, 0);
  }
#ifdef HAVE_WAIT_ASYNC_BUILTIN
  __builtin_amdgcn_s_wait_asynccnt(0);
#else
  asm volatile("s_wait_asynccnt 0" ::: "memory");
#endif
#else
#pragma unroll
  for (int i = 0; i < EPT / 4; ++i) {
    const int idx = (i * BLOCK + tid) * 4;
    *reinterpret_cast<float4*>(&rowbuf[idx]) =
        *reinterpret_cast<const float4*>(rp + idx);
  }
#endif
  __syncthreads();

  // ---- Pass 1: row max (from LDS) ----
  float m = -3.402823466e+38f;
#pragma unroll
  for (int i = 0; i < EPT; ++i)
    m = fmaxf(m, rowbuf[i * BLOCK + tid]);
  m = wave_max32(m);
  const int wave = tid >> 5;
  if ((tid & 31) == 0) wred[wave] = m;
  __syncthreads();
  if (tid == 0) {
    float t = wred[0];
#pragma unroll
    for (int i = 1; i < 8; ++i) t = fmaxf(t, wred[i]);
    bcast[0] = t;
  }
  __syncthreads();
  m = bcast[0];

  // ---- Pass 2: total sum of exp(x - m), WMMA-reduced per wave ----
  float s = 0.0f;
#pragma unroll
  for (int i = 0; i < EPT; ++i)
    s += __expf(rowbuf[i * BLOCK + tid] - m);
  s = wave_sum32_wmma(s);
  if ((tid & 31) == 0) wred[wave] = s;
  __syncthreads();
  if (tid == 0) {
    float t = 0.0f;
#pragma unroll
    for (int i = 0; i < 8; ++i) t += wred[i];
    bcast[1] = t;
  }
  // ---- Gather positives from the LDS-resident row ----
  if (tid < P_DIM) {
    const long long lab = labels[(size_t)b * P_DIM + tid];
    const float x = rowbuf[(int)lab];
    posx[tid] = x;
    pose[tid] = __expf(x - m);
  }
  __syncthreads();

  // ---- Per-row loss: lse over {negatives + this positive}, minus pos logit
  if (tid == 0) {
    const float total = bcast[1];
    float spos = 0.0f;
#pragma unroll
    for (int p = 0; p < P_DIM; ++p) spos += pose[p];
    float neg = total - spos; // sum over negatives only
    if (neg < 0.0f) neg = 0.0f;
    float loss = 0.0f;
#pragma unroll
    for (int p = 0; p < P_DIM; ++p)
      loss += m + logf(neg + pose[p]) - posx[p];
    row_loss[b] = loss;
  }
}

// Deterministic final mean over per-row losses (no float atomics).
__global__ void __launch_bounds__(BLOCK)
mlce_reduce_kernel(const float* __restrict__ row_loss, float* __restrict__ out,
                   int n) {
  __shared__ float sm[BLOCK];
  float s = 0.0f;
  for (int i = threadIdx.x; i < n; i += BLOCK) s += row_loss[i];
  sm[threadIdx.x] = s;
  __syncthreads();
#pragma unroll
  for (int off = BLOCK / 2; off > 0; off >>= 1) {
    if (threadIdx.x < off) sm[threadIdx.x] += sm[threadIdx.x + off];
    __syncthreads();
  }
  if (threadIdx.x == 0) out[0] = sm[0] / ((float)n * (float)P_DIM);
}

extern "C" void kernel_launch(void* const* d_in, const int* in_sizes, int n_in,
                              void* d_out, int out_size, void* d_ws,
                              size_t ws_size, hipStream_t stream) {
  const float* pred = (const float*)d_in[0];           // [B, C] float32
  const long long* labels = (const long long*)d_in[1]; // [B, P] int64
  float* out = (float*)d_out;                          // scalar float32
  float* row_loss = (float*)d_ws;                      // B floats of scratch

  const int nrows = in_sizes[1] / P_DIM; // B = 2048

  mlce_row_kernel<<<nrows, BLOCK, 0, stream>>>(pred, labels, row_loss);
  mlce_reduce_kernel<<<1, BLOCK, 0, stream>>>(row_loss, out, nrows);
}